// CosineWeights_18262200942963
// MI455X (gfx1250) — compile-verified
//
#include <hip/hip_runtime.h>
#include <math.h>

// CDNA5 / gfx1250, wave32. Cosine-similarity content addressing:
//   scores = keys @ memory^T ; normed = scores/(|m||k|+eps) ; softmax(normed*softplus(s))
// Bandwidth-bound: read memory (512MB) exactly once, fuse everything into that
// pass using V_WMMA_F32_16X16X4_F32; softmax is a second pass over 32MB.

typedef float v2f __attribute__((ext_vector_type(2)));
typedef float v8f __attribute__((ext_vector_type(8)));

#define B_   64
#define M_   16384
#define W_   128
#define H_   8
#define EPSF 1e-6f

#define KSTRIDE 132   // LDS row stride for keys (pad 128 -> 132 to spread banks)

// ---------------------------------------------------------------------------
// K0: per (b,h) key L2 norm and softplus(strength). 512 values total.
// ---------------------------------------------------------------------------
__global__ void key_stats_kernel(const float* __restrict__ keys,
                                 const float* __restrict__ strengths,
                                 float* __restrict__ kn,
                                 float* __restrict__ sp) {
    int idx = blockIdx.x * blockDim.x + threadIdx.x;
    if (idx >= B_ * H_) return;
    const float* kp = keys + (size_t)idx * W_;
    float s = 0.f;
    #pragma unroll 4
    for (int w = 0; w < W_; ++w) { float v = kp[w]; s += v * v; }
    kn[idx] = sqrtf(s);
    float x = strengths[idx];
    sp[idx] = fmaxf(x, 0.f) + log1pf(__expf(-fabsf(x)));   // stable softplus
}

// ---------------------------------------------------------------------------
// K1: fused scores + cosine normalization + strength scaling.
// One wave computes a 16(head-pad) x 16(memory-row) tile via 32x
// v_wmma_f32_16x16x4_f32 accumulations over W=128.
//
// WMMA f32 16x16x4 operand layout assumed (ISA 7.12.2):
//   A (16x4, M=head): lane L<16 -> row L, K={0,1}; lane L>=16 -> row L-16, K={2,3}
//   B (4x16,  N=mem): lane L<16 -> col L, K={0,1}; lane L>=16 -> col L-16, K={2,3}
//   C/D: VGPR j: lanes 0-15 -> M=j, N=lane ; lanes 16-31 -> M=j+8, N=lane-16
// Only heads 0..7 are real; rows 8..15 of A are zero (LDS zero-padded), so
// the D VGPRs of lanes 0..15 carry all valid output (heads 0..7).
// ---------------------------------------------------------------------------
__global__ void scores_kernel(const float* __restrict__ mem,
                              const float* __restrict__ keys,
                              const float* __restrict__ kn,
                              const float* __restrict__ sp,
                              float* __restrict__ out) {
    __shared__ float kld[16 * KSTRIDE];

    const int b   = blockIdx.y;
    const int tid = threadIdx.x;

    // Stage keys[b] (8x128) into LDS, zero-pad heads 8..15.
    for (int i = tid; i < 16 * W_; i += blockDim.x) {
        int row = i / W_, col = i % W_;
        kld[row * KSTRIDE + col] =
            (row < H_) ? keys[((size_t)b * H_ + row) * W_ + col] : 0.f;
    }
    __syncthreads();

    const int wave = tid >> 5;
    const int lane = tid & 31;
    const int m0   = (blockIdx.x * (blockDim.x >> 5) + wave) * 16;
    const int r    = lane & 15;          // mem row in tile (B col) / key row (A)
    const int ksel = (lane >> 4) * 2;    // which K pair of the 4-wide chunk

    const float* mrow = mem + ((size_t)b * M_ + (m0 + r)) * W_ + ksel;
    const float* krow = kld + r * KSTRIDE + ksel;

    v8f   acc = {};
    float sq  = 0.f;                     // partial sq-sum of this memory row

    #pragma unroll
    for (int k = 0; k < W_ / 4; ++k) {
        v2f a, bm;
        a.x  = krow[4 * k + 0];
        a.y  = krow[4 * k + 1];
        bm.x = mrow[4 * k + 0];
        bm.y = mrow[4 * k + 1];
        sq  += bm.x * bm.x + bm.y * bm.y;
        acc = __builtin_amdgcn_wmma_f32_16x16x4_f32(
                  /*neg_a=*/false, a, /*neg_b=*/false, bm,
                  /*c_mod=*/(short)0, acc, /*reuse_a=*/false, /*reuse_b=*/false);
    }

    // Lanes L and L+16 hold complementary K-halves of the same memory row.
    sq += __shfl_xor(sq, 16, 32);
    const float mnorm = sqrtf(sq);

    // Valid output lives in lanes 0..15: head=j, mem col = m0+lane.
    if (lane < 16) {
        const int m = m0 + lane;
        #pragma unroll
        for (int j = 0; j < H_; ++j) {
            const float denom = mnorm * kn[b * H_ + j] + EPSF;
            out[((size_t)b * H_ + j) * M_ + m] = acc[j] / denom * sp[b * H_ + j];
        }
    }
}

// ---------------------------------------------------------------------------
// K2: in-place softmax over M=16384 per (b,h) row. One 256-thread block per
// row; 64 values per thread stay in registers (one read + one write of 32MB).
// ---------------------------------------------------------------------------
__global__ void softmax_kernel(float* __restrict__ out) {
    const int tid = threadIdx.x;
    float* row = out + (size_t)blockIdx.x * M_;

    float v[64];
    float lmax = -3.402823466e38f;
    #pragma unroll
    for (int j = 0; j < 64; ++j) {
        v[j] = row[tid + j * 256];
        lmax = fmaxf(lmax, v[j]);
    }

    __shared__ float redmax[8];
    __shared__ float redsum[8];

    #pragma unroll
    for (int o = 16; o > 0; o >>= 1) lmax = fmaxf(lmax, __shfl_xor(lmax, o, 32));
    if ((tid & 31) == 0) redmax[tid >> 5] = lmax;
    __syncthreads();
    float gmax = redmax[0];
    #pragma unroll
    for (int i = 1; i < 8; ++i) gmax = fmaxf(gmax, redmax[i]);

    float lsum = 0.f;
    #pragma unroll
    for (int j = 0; j < 64; ++j) {
        v[j] = __expf(v[j] - gmax);
        lsum += v[j];
    }
    #pragma unroll
    for (int o = 16; o > 0; o >>= 1) lsum += __shfl_xor(lsum, o, 32);
    if ((tid & 31) == 0) redsum[tid >> 5] = lsum;
    __syncthreads();
    float gsum = 0.f;
    #pragma unroll
    for (int i = 0; i < 8; ++i) gsum += redsum[i];

    const float inv = 1.f / gsum;
    #pragma unroll
    for (int j = 0; j < 64; ++j) row[tid + j * 256] = v[j] * inv;
}

// ---------------------------------------------------------------------------
extern "C" void kernel_launch(void* const* d_in, const int* in_sizes, int n_in,
                              void* d_out, int out_size, void* d_ws, size_t ws_size,
                              hipStream_t stream) {
    const float* memory    = (const float*)d_in[0];   // [B, M, W]
    const float* keys      = (const float*)d_in[1];   // [B, H, W]
    const float* strengths = (const float*)d_in[2];   // [B, H]
    float*       out       = (float*)d_out;           // [B, H, M]

    float* kn = (float*)d_ws;           // [B*H] key norms
    float* sp = kn + B_ * H_;           // [B*H] softplus(strength)

    // K0: key stats (512 (b,h) pairs)
    key_stats_kernel<<<dim3((B_ * H_ + 255) / 256), dim3(256), 0, stream>>>(
        keys, strengths, kn, sp);

    // K1: fused WMMA scores. 8 waves/block, each wave = one 16-row m-tile.
    dim3 g1(M_ / (16 * 8), B_);
    scores_kernel<<<g1, dim3(256), 0, stream>>>(memory, keys, kn, sp, out);

    // K2: softmax per (b,h) row.
    softmax_kernel<<<dim3(B_ * H_), dim3(256), 0, stream>>>(out);
}